// MultiBandSpectralConv_850403525021
// MI455X (gfx1250) — compile-verified
//
#include <hip/hip_runtime.h>

// MI455X / gfx1250, wave32. Fused 5-level Db4 DWT, one workgroup per (b,c) row.
// HBM-bound: ~268MB in + ~268MB out => ~23us roofline at 23.3 TB/s.
// Conv-as-Toeplitz-GEMM via V_WMMA_F32_16X16X4_F32 (full fp32):
//   A(16x4)  = 16 output positions x 4 taps of the padded signal
//   B(4x16)  = taps x {col0=low filter, col1=high filter, cols2..15 = 0}
//   D(16x16) = col0 -> low band, col1 -> high band; two WMMAs cover 8 taps.

typedef __attribute__((ext_vector_type(2))) float v2f;
typedef __attribute__((ext_vector_type(8))) float v8f;

namespace {
constexpr int    SLEN  = 16384;
constexpr int    NROWS = 4096;                            // 64*64
constexpr int    NT    = 256;                             // 8 wave32 waves
constexpr size_t OFF_FD = 0;
constexpr size_t OFF_FT = OFF_FD + (size_t)NROWS * 512;   // 2,097,152
constexpr size_t OFF_FA = OFF_FT + (size_t)NROWS * 512;   // 4,194,304
constexpr size_t OFF_FB = OFF_FA + (size_t)NROWS * 1024;  // 8,388,608
constexpr size_t OFF_FG = OFF_FB + (size_t)NROWS * 2048;  // 16,777,216
constexpr size_t OFF_FR = OFF_FG + (size_t)NROWS * 4096;  // 33,554,432
}

__global__ __launch_bounds__(NT) void dwt_db4_rows(
    const float* __restrict__ x,
    const float* __restrict__ flo,
    const float* __restrict__ fhi,
    float* __restrict__ out)
{
  // Ping-pong padded LDS buffers: data x[i] at buf[8+i], reflect prefix in [1..7]
  // so every level computes out[n] = sum_k f[k] * buf[1 + 2n + k].
  __shared__ __align__(16) float buf0[8 + 8192];   // levels 1,3,5 low output
  __shared__ __align__(16) float buf1[8 + 4096];   // levels 2,4 low output

  const int row  = blockIdx.x;
  const int tid  = threadIdx.x;
  const int lane = tid & 31;
  const int wav  = tid >> 5;           // 0..7
  const int ncol = lane & 15;          // N column (also M row id within half)
  const int kr   = (lane >> 4) << 1;   // K sub-block: 0 for lanes 0-15, 2 for 16-31

  const float* __restrict__ xrow = x + (size_t)row * SLEN;
  const size_t frbase = OFF_FR + (size_t)row * 8192;

  // B matrices (4x16, KxN). VGPR0 holds K={0|2} rows, VGPR1 holds K={1|3}.
  // Column 0 = low-pass filter, column 1 = high-pass filter, others zero.
  float b1x = 0.f, b1y = 0.f, b2x = 0.f, b2y = 0.f;
  if (ncol == 0)      { b1x = flo[kr]; b1y = flo[kr + 1]; b2x = flo[kr + 4]; b2y = flo[kr + 5]; }
  else if (ncol == 1) { b1x = fhi[kr]; b1y = fhi[kr + 1]; b2x = fhi[kr + 4]; b2y = fhi[kr + 5]; }
  v2f B1; B1.x = b1x; B1.y = b1y;      // taps 0..3
  v2f B2; B2.x = b2x; B2.y = b2y;      // taps 4..7

  float* inb  = nullptr;
  float* outb = buf0;
  int L = SLEN;

  for (int lev = 1; lev <= 5; ++lev) {
    const int Lo = L >> 1;
    const int T  = Lo >> 4;            // 16-output tiles per level

    // High-band destinations (level-1 high is unused by the reference).
    float* hA = nullptr; float* hB = nullptr;
    if      (lev == 2) { hA = out + OFF_FG + (size_t)row * 4096; hB = out + frbase + 4096; }
    else if (lev == 3) { hA = out + OFF_FB + (size_t)row * 2048; hB = out + frbase + 2048; }
    else if (lev == 4) { hA = out + OFF_FA + (size_t)row * 1024; hB = out + frbase + 1024; }
    else if (lev == 5) { hA = out + OFF_FT + (size_t)row * 512;  hB = out + frbase + 512;  }

    for (int t = wav; t < T; t += NT / 32) {
      const int m0 = t << 4;
      const int t0 = ((m0 + ncol) << 1) + kr;   // padded-signal index of tap 0

      // A matrix (16x4 Toeplitz): VGPR0 = xp[t0], VGPR1 = xp[t0+1] (K split by kr).
      v2f A1, A2;
      if (lev == 1) {
        // Read x directly from global; reflect pad via |t-7|.
        int i0 = t0 - 7; if (i0 < 0) i0 = -i0;
        int i1 = t0 - 6; if (i1 < 0) i1 = -i1;
        int i4 = t0 - 3; if (i4 < 0) i4 = -i4;
        int i5 = t0 - 2; if (i5 < 0) i5 = -i5;
        A1.x = xrow[i0]; A1.y = xrow[i1];
        A2.x = xrow[i4]; A2.y = xrow[i5];
      } else {
        const float* p = inb + 1 + t0;
        A1.x = p[0]; A1.y = p[1];
        A2.x = p[4]; A2.y = p[5];
      }

      v8f acc = {};
      acc = __builtin_amdgcn_wmma_f32_16x16x4_f32(false, A1, false, B1,
                                                  (short)0, acc, false, false);
      acc = __builtin_amdgcn_wmma_f32_16x16x4_f32(false, A2, false, B2,
                                                  (short)0, acc, false, false);

      // D layout: VGPR r, lanes 0-15 -> D[r, lane]; lanes 16-31 -> D[r+8, lane-16].
      const int half8 = (lane >> 4) << 3;       // 0 or 8
      if (ncol == 0) {                          // lanes 0,16: low band, 8 rows each
        float* p = outb + 8 + m0 + half8;
        float4 l0; l0.x = acc[0]; l0.y = acc[1]; l0.z = acc[2]; l0.w = acc[3];
        float4 l1; l1.x = acc[4]; l1.y = acc[5]; l1.z = acc[6]; l1.w = acc[7];
        reinterpret_cast<float4*>(p)[0] = l0;
        reinterpret_cast<float4*>(p)[1] = l1;
      }
      if (lev >= 2 && ncol == 1) {              // lanes 1,17: high band
        const int off = m0 + half8;
        float4 h0; h0.x = acc[0]; h0.y = acc[1]; h0.z = acc[2]; h0.w = acc[3];
        float4 h1; h1.x = acc[4]; h1.y = acc[5]; h1.z = acc[6]; h1.w = acc[7];
        reinterpret_cast<float4*>(hA + off)[0] = h0;
        reinterpret_cast<float4*>(hA + off)[1] = h1;
        reinterpret_cast<float4*>(hB + off)[0] = h0;
        reinterpret_cast<float4*>(hB + off)[1] = h1;
      }
    }

    __syncthreads();
    // Reflect prefix for the next level: buf[8-j] = low[j], j=1..7.
    if (tid >= 1 && tid <= 7) outb[8 - tid] = outb[8 + tid];
    __syncthreads();

    inb  = outb;
    outb = (outb == buf0) ? buf1 : buf0;
    L = Lo;
  }

  // Level-5 low = Fd (512 floats, sits in inb after the final swap).
  for (int i = tid; i < 512; i += NT) {
    const float v = inb[8 + i];
    out[OFF_FD + (size_t)row * 512 + i] = v;
    out[frbase + i] = v;
  }
}

extern "C" void kernel_launch(void* const* d_in, const int* in_sizes, int n_in,
                              void* d_out, int out_size, void* d_ws, size_t ws_size,
                              hipStream_t stream) {
  const float* x   = (const float*)d_in[0];
  const float* flo = (const float*)d_in[1];
  const float* fhi = (const float*)d_in[2];
  float* out = (float*)d_out;
  (void)in_sizes; (void)n_in; (void)out_size; (void)d_ws; (void)ws_size;

  dwt_db4_rows<<<NROWS, NT, 0, stream>>>(x, flo, fhi, out);
}